// CTPINNLoss_10651518894371
// MI455X (gfx1250) — compile-verified
//
#include <hip/hip_runtime.h>
#include <hip/hip_bf16.h>
#include <math.h>

// CT-PINN loss for MI455X (gfx1250, wave32).
// Kernel 1: Radon line integration, 1 ray per thread, 2 blocks per angle.
//           Branch-free clamped bilinear taps (med3+load+cndmask, no exec ladders),
//           global_prefetch_b8 ahead along each ray. MSE partial per block.
// Kernel 2: TV partials, idiv-free index math, coalesced unconditional loads.
// Kernel 3: finalize. Cross-lane sums via v_wmma_f32_16x16x4_f32 (ones-B trick).

typedef __attribute__((ext_vector_type(2))) float v2f;
typedef __attribute__((ext_vector_type(8))) float v8f;

#define IMG_W 512
#define IMG_H 512
#define TV_BLOCKS 256

// ---- wave32 sum via WMMA f32 16x16x4 with B == ones (B layout-independent) ----
// A layout (32-bit 16x4): lanes 0-15 -> {K0,K1}, lanes 16-31 -> {K2,K3}, M = lane%16.
// a = {v, 0} => D[m, n] = v[m] + v[m+16]. Summing the 8 D VGPRs per lane gives the
// row-half sums; one shfl_xor(16) completes the 32-lane total. EXEC all-ones at all
// call sites (invoked only after reconvergence with full 256-thread blocks).
__device__ __forceinline__ float wave_sum_wmma(float v) {
  v2f a; a[0] = v;    a[1] = 0.0f;
  v2f b; b[0] = 1.0f; b[1] = 1.0f;
  v8f c = {};
  c = __builtin_amdgcn_wmma_f32_16x16x4_f32(false, a, false, b, (short)0, c, false, false);
  float s = c[0] + c[1] + c[2] + c[3] + c[4] + c[5] + c[6] + c[7];
  s += __shfl_xor(s, 16, 32);
  return s;
}

// Block (256-thread, 8-wave) sum; result valid on thread 0.
__device__ __forceinline__ float block_sum(float v) {
  __shared__ float lds[8];
  float ws = wave_sum_wmma(v);
  const int lane = threadIdx.x & 31;
  const int wid  = threadIdx.x >> 5;
  if (lane == 0) lds[wid] = ws;
  __syncthreads();
  float r = 0.0f;
  if (threadIdx.x == 0) {
    #pragma unroll
    for (int i = 0; i < 8; ++i) r += lds[i];
  }
  __syncthreads();  // allow lds reuse by a subsequent block_sum
  return r;
}

// Branch-free zero-padded tap: clamp (med3), unconditional load, post-select.
__device__ __forceinline__ float samp(const float* __restrict__ img, int y, int x) {
  const int xc = min(max(x, 0), IMG_W - 1);
  const int yc = min(max(y, 0), IMG_H - 1);
  const float v = img[yc * IMG_W + xc];
  const bool in = (((unsigned)x < (unsigned)IMG_W) & ((unsigned)y < (unsigned)IMG_H));
  return in ? v : 0.0f;
}

// gridDim.x = 2*A; block handles 256 detector bins of one angle, 1 ray/thread.
__global__ __launch_bounds__(256) void radon_mse_kernel(
    const float* __restrict__ img, const float* __restrict__ sino,
    const float* __restrict__ angles, float* __restrict__ partial) {
  const int a = blockIdx.x >> 1;
  const int w = ((blockIdx.x & 1) << 8) + (int)threadIdx.x;
  const float ang = angles[a];
  const float si = sinf(ang);
  const float co = cosf(ang);

  // Normalized detector coordinate and first-row sample position.
  const float xw  = (2.0f * (float)w + 1.0f) * (1.0f / (float)IMG_W) - 1.0f;
  const float yh0 = (1.0f / (float)IMG_H) - 1.0f;
  const float gx = co * xw - si * yh0;
  const float gy = si * xw + co * yh0;
  float ix = fmaf(gx + 1.0f, 0.5f * (float)IMG_W, -0.5f);
  float iy = fmaf(gy + 1.0f, 0.5f * (float)IMG_H, -0.5f);
  // Unit-step ray direction (W==H): exactly one pixel per source row.
  const float dix = -si;
  const float diy = co;

  float acc = 0.0f;
  for (int h0 = 0; h0 < IMG_H; h0 += 16) {
    // Prefetch ~24 steps ahead along the ray (gfx1250 global_prefetch_b8).
    {
      const float pfx = fmaf(24.0f, dix, ix);
      const float pfy = fmaf(24.0f, diy, iy);
      const int px = min(max((int)pfx, 0), IMG_W - 1);
      const int py = min(max((int)pfy, 0), IMG_H - 1);
      __builtin_prefetch(img + py * IMG_W + px, 0, 1);
    }
    #pragma unroll
    for (int hh = 0; hh < 16; ++hh) {
      const float xf = floorf(ix);
      const float yf = floorf(iy);
      const float wx = ix - xf;
      const float wy = iy - yf;
      const int x0 = (int)xf;
      const int y0 = (int)yf;
      const float v00 = samp(img, y0,     x0);
      const float v01 = samp(img, y0,     x0 + 1);
      const float v10 = samp(img, y0 + 1, x0);
      const float v11 = samp(img, y0 + 1, x0 + 1);
      const float top = fmaf(wx, v01 - v00, v00);
      const float bot = fmaf(wx, v11 - v10, v10);
      acc += fmaf(wy, bot - top, top);
      ix += dix;
      iy += diy;
    }
  }
  const float d = acc - sino[a * IMG_W + w];
  const float bs = block_sum(d * d);
  if (threadIdx.x == 0) partial[blockIdx.x] = bs;
}

// TV over 511x511; iterate the 511x512 row-major range so r = i>>9, c = i&511
// (no integer divide). All loads in-bounds and coalesced; c==511 masked out.
__global__ __launch_bounds__(256) void tv_kernel(
    const float* __restrict__ pred, float* __restrict__ partial) {
  const int N = (IMG_H - 1) * IMG_W;  // 511 * 512
  float local = 0.0f;
  for (int i = blockIdx.x * blockDim.x + threadIdx.x; i < N;
       i += gridDim.x * blockDim.x) {
    const int c = i & (IMG_W - 1);
    const float pc = pred[i];
    const float dx = pred[i + 1] - pc;       // i+1 < 512*512 always here
    const float dy = pred[i + IMG_W] - pc;   // r < 511 => in bounds
    const float t = sqrtf(fmaf(dx, dx, fmaf(dy, dy, 1e-8f)));
    local += (c < IMG_W - 1) ? t : 0.0f;
  }
  const float bs = block_sum(local);
  if (threadIdx.x == 0) partial[blockIdx.x] = bs;
}

__global__ __launch_bounds__(256) void finalize_kernel(
    const float* __restrict__ partial, float* __restrict__ out, int nA) {
  const int nProj = 2 * nA;  // radon partial count (2 blocks per angle)
  float v1 = 0.0f;
  for (int j = threadIdx.x; j < nProj; j += 256) v1 += partial[j];
  const float proj = block_sum(v1);
  const float v2 = partial[nProj + threadIdx.x];  // TV_BLOCKS == 256 entries
  const float tv = block_sum(v2);
  if (threadIdx.x == 0) {
    const float proj_mean = proj / ((float)nA * (float)IMG_W);
    const float tv_mean   = tv / ((float)(IMG_H - 1) * (float)(IMG_W - 1));
    out[0] = proj_mean + 0.01f * tv_mean;
  }
}

extern "C" void kernel_launch(void* const* d_in, const int* in_sizes, int n_in,
                              void* d_out, int out_size, void* d_ws, size_t ws_size,
                              hipStream_t stream) {
  const float* pred   = (const float*)d_in[0];  // [1,512,512]
  const float* sino   = (const float*)d_in[1];  // [1,A,512]
  const float* angles = (const float*)d_in[2];  // [A]
  float* out = (float*)d_out;
  float* partial = (float*)d_ws;                // [2A + TV_BLOCKS] floats
  const int A = in_sizes[2];                    // 180

  radon_mse_kernel<<<2 * A, 256, 0, stream>>>(pred, sino, angles, partial);
  tv_kernel<<<TV_BLOCKS, 256, 0, stream>>>(pred, partial + 2 * A);
  finalize_kernel<<<1, 256, 0, stream>>>(partial, out, A);
}